// GAT_NET_56891136803142
// MI455X (gfx1250) — compile-verified
//
#include <hip/hip_runtime.h>
#include <hip/hip_bf16.h>
#include <math.h>

// ---------------- problem constants (from reference) ----------------
#define N_NODES 50000
#define N_EDGES 800000
#define E_TOT   (N_EDGES + N_NODES)   // reference appends self-loops
#define FIN     256
#define HEADS   4
#define HIDC    64
#define C1      (HEADS * HIDC)        // 256 = layer-1 output width
#define CLS     8
#define NEG_SLOPE 0.2f

typedef __attribute__((ext_vector_type(2))) float v2f;
typedef __attribute__((ext_vector_type(8))) float v8f;

__device__ __forceinline__ float leaky(float x) { return x > 0.f ? x : NEG_SLOPE * x; }

__device__ __forceinline__ void edge_sd(const int* ei, int e, int& s, int& d) {
    if (e < N_EDGES) { s = ei[e]; d = ei[N_EDGES + e]; }
    else             { s = e - N_EDGES; d = s; }          // self loop
}

// float atomic max via sign-aware integer atomics (monotone bit encoding)
__device__ __forceinline__ void atomicMaxF(float* addr, float val) {
    if (val >= 0.f) atomicMax((int*)addr, __float_as_int(val));
    else            atomicMin((unsigned int*)addr, __float_as_uint(val));
}

// ---------------- init: zero accumulators, -inf maxima ----------------
__global__ void init_kernel(float* acc1, float* m1, float* s1,
                            float* m2, float* s2, float* acc2) {
    int i = blockIdx.x * blockDim.x + threadIdx.x;          // covers N*C1
    if (i < N_NODES * C1) acc1[i] = 0.f;
    if (i < N_NODES * HEADS) { m1[i] = -INFINITY; s1[i] = 0.f; }
    if (i < N_NODES) { m2[i] = -INFINITY; s2[i] = 0.f; }
    if (i < N_NODES * CLS) acc2[i] = 0.f;
}

// ---------------- GEMM1: h1[N,256] = x[N,256] @ W1[256,256] ----------------
// one wave per 16x64 output strip: A fragment loaded once per k-step and
// reused across 4 column tiles (4 accumulators, 4 WMMAs per k-step).
__global__ void gemm1_wmma(const float* __restrict__ x,
                           const float* __restrict__ W1,
                           float* __restrict__ h1) {
    const int lane = threadIdx.x;            // 0..31, wave32
    const int mt  = blockIdx.x;              // 3125 row tiles (N divisible by 16)
    const int ntb = blockIdx.y;              // 4 strips of 64 columns
    const int r16  = lane & 15;
    const int half = lane >> 4;              // 0: K pair {0,1}, 1: K pair {2,3}
    const int arow  = mt * 16 + r16;
    const int cbase = ntb * 64 + r16;

    v8f acc0 = {}, acc1 = {}, acc2 = {}, acc3 = {};
#pragma unroll 8
    for (int k = 0; k < FIN; k += 4) {
        const int kk = k + half * 2;
        v2f a;
        a.x = x[arow * FIN + kk];
        a.y = x[arow * FIN + kk + 1];
        const float* w0 = W1 + (kk)     * C1 + cbase;
        const float* w1 = W1 + (kk + 1) * C1 + cbase;
        v2f b0 = { w0[0],  w1[0]  };
        v2f b1 = { w0[16], w1[16] };
        v2f b2 = { w0[32], w1[32] };
        v2f b3 = { w0[48], w1[48] };
        acc0 = __builtin_amdgcn_wmma_f32_16x16x4_f32(false, a, false, b0, (short)0, acc0, false, false);
        acc1 = __builtin_amdgcn_wmma_f32_16x16x4_f32(false, a, false, b1, (short)0, acc1, false, false);
        acc2 = __builtin_amdgcn_wmma_f32_16x16x4_f32(false, a, false, b2, (short)0, acc2, false, false);
        acc3 = __builtin_amdgcn_wmma_f32_16x16x4_f32(false, a, false, b3, (short)0, acc3, false, false);
    }
    // D layout: v[i] -> row mt*16 + half*8 + i, col = lane&15 within each tile
    const int rbase = mt * 16 + half * 8;
#pragma unroll
    for (int i = 0; i < 8; ++i) {
        const int ro = (rbase + i) * C1 + ntb * 64 + r16;
        h1[ro]      = acc0[i];
        h1[ro + 16] = acc1[i];
        h1[ro + 32] = acc2[i];
        h1[ro + 48] = acc3[i];
    }
}

// ---------------- per-(node,head) attention coefficients ----------------
__global__ void alpha1_kernel(const float* __restrict__ h1,
                              const float* __restrict__ att_src,
                              const float* __restrict__ att_dst,
                              float* __restrict__ a1s, float* __restrict__ a1d) {
    int idx = blockIdx.x * blockDim.x + threadIdx.x;        // (n,h)
    if (idx >= N_NODES * HEADS) return;
    const int n = idx >> 2, h = idx & 3;
    const float* hr = h1 + n * C1 + h * HIDC;
    const float* as = att_src + h * HIDC;
    const float* ad = att_dst + h * HIDC;
    float ss = 0.f, dd = 0.f;
#pragma unroll 8
    for (int c = 0; c < HIDC; ++c) { float v = hr[c]; ss += v * as[c]; dd += v * ad[c]; }
    a1s[idx] = ss; a1d[idx] = dd;
}

// ---------------- edge pass 1: segment max of leaky logits ----------------
__global__ void edge_max1(const int* __restrict__ ei,
                          const float* __restrict__ a1s, const float* __restrict__ a1d,
                          float* __restrict__ m1) {
    int e = blockIdx.x * blockDim.x + threadIdx.x;
    if (e >= E_TOT) return;
    int s, d; edge_sd(ei, e, s, d);
    const float4 es = *(const float4*)(a1s + s * HEADS);    // 16B aligned
    const float4 ed = *(const float4*)(a1d + d * HEADS);
    atomicMaxF(&m1[d * HEADS + 0], leaky(es.x + ed.x));
    atomicMaxF(&m1[d * HEADS + 1], leaky(es.y + ed.y));
    atomicMaxF(&m1[d * HEADS + 2], leaky(es.z + ed.z));
    atomicMaxF(&m1[d * HEADS + 3], leaky(es.w + ed.w));
}

// ---------------- edge pass 2: segment sum of exp ----------------
__global__ void edge_sum1(const int* __restrict__ ei,
                          const float* __restrict__ a1s, const float* __restrict__ a1d,
                          const float* __restrict__ m1, float* __restrict__ s1) {
    int e = blockIdx.x * blockDim.x + threadIdx.x;
    if (e >= E_TOT) return;
    int s, d; edge_sd(ei, e, s, d);
    const float4 es = *(const float4*)(a1s + s * HEADS);
    const float4 ed = *(const float4*)(a1d + d * HEADS);
    const float4 mm = *(const float4*)(m1  + d * HEADS);
    atomicAdd(&s1[d * HEADS + 0], expf(leaky(es.x + ed.x) - mm.x));
    atomicAdd(&s1[d * HEADS + 1], expf(leaky(es.y + ed.y) - mm.y));
    atomicAdd(&s1[d * HEADS + 2], expf(leaky(es.z + ed.z) - mm.z));
    atomicAdd(&s1[d * HEADS + 3], expf(leaky(es.w + ed.w) - mm.w));
}

// ---------------- edge pass 3: weighted scatter of source rows ----------------
// one 256-thread block per edge; thread c handles channel c (head = c/64)
__global__ void edge_acc1(const int* __restrict__ ei,
                          const float* __restrict__ a1s, const float* __restrict__ a1d,
                          const float* __restrict__ m1, const float* __restrict__ s1,
                          const float* __restrict__ h1, float* __restrict__ acc1) {
    const int e = blockIdx.x;
    const int c = threadIdx.x;
    const int h = c >> 6;
    int s, d; edge_sd(ei, e, s, d);
    float lg = leaky(a1s[s * HEADS + h] + a1d[d * HEADS + h]);
    float alpha = expf(lg - m1[d * HEADS + h]) / s1[d * HEADS + h];
    atomicAdd(&acc1[d * C1 + c], alpha * h1[s * C1 + c]);
}

// ---------------- bias + ReLU (in place, acc1 becomes layer-2 features) ---
__global__ void bias_relu(float* __restrict__ acc1, const float* __restrict__ b1) {
    int i = blockIdx.x * blockDim.x + threadIdx.x;
    if (i >= N_NODES * C1) return;
    float v = acc1[i] + b1[i & (C1 - 1)];
    acc1[i] = v > 0.f ? v : 0.f;
}

// ---------------- GEMM2: g2[N,8] = feat[N,256] @ W2[256,8] ----------------
// W2 zero-padded to 256x16 in LDS; one wave per 16-row tile
__global__ void gemm2_wmma(const float* __restrict__ feat,
                           const float* __restrict__ W2,
                           float* __restrict__ g2) {
    __shared__ float Bs[FIN * 16];
    const int lane = threadIdx.x;
    for (int i = lane; i < FIN * 16; i += 32) {
        int k = i >> 4, c = i & 15;
        Bs[i] = (c < CLS) ? W2[k * CLS + c] : 0.f;
    }
    __syncthreads();

    const int mt = blockIdx.x;
    const int r16  = lane & 15;
    const int half = lane >> 4;
    const int arow = mt * 16 + r16;

    v8f acc = {};
#pragma unroll 8
    for (int k = 0; k < FIN; k += 4) {
        const int kk = k + half * 2;
        v2f a, b;
        a.x = feat[arow * FIN + kk];
        a.y = feat[arow * FIN + kk + 1];
        b.x = Bs[(kk)     * 16 + r16];
        b.y = Bs[(kk + 1) * 16 + r16];
        acc = __builtin_amdgcn_wmma_f32_16x16x4_f32(
            false, a, false, b, (short)0, acc, false, false);
    }
    const int rbase = mt * 16 + half * 8;
    if (r16 < CLS) {
#pragma unroll
        for (int i = 0; i < 8; ++i)
            g2[(rbase + i) * CLS + r16] = acc[i];
    }
}

// ---------------- layer-2 attention coefficients (1 head, 8 ch) ----------
__global__ void alpha2_kernel(const float* __restrict__ g2,
                              const float* __restrict__ att_src,
                              const float* __restrict__ att_dst,
                              float* __restrict__ a2s, float* __restrict__ a2d) {
    int n = blockIdx.x * blockDim.x + threadIdx.x;
    if (n >= N_NODES) return;
    float ss = 0.f, dd = 0.f;
#pragma unroll
    for (int c = 0; c < CLS; ++c) { float v = g2[n * CLS + c]; ss += v * att_src[c]; dd += v * att_dst[c]; }
    a2s[n] = ss; a2d[n] = dd;
}

__global__ void edge_max2(const int* __restrict__ ei,
                          const float* __restrict__ a2s, const float* __restrict__ a2d,
                          float* __restrict__ m2) {
    int e = blockIdx.x * blockDim.x + threadIdx.x;
    if (e >= E_TOT) return;
    int s, d; edge_sd(ei, e, s, d);
    atomicMaxF(&m2[d], leaky(a2s[s] + a2d[d]));
}

__global__ void edge_sum2(const int* __restrict__ ei,
                          const float* __restrict__ a2s, const float* __restrict__ a2d,
                          const float* __restrict__ m2, float* __restrict__ s2) {
    int e = blockIdx.x * blockDim.x + threadIdx.x;
    if (e >= E_TOT) return;
    int s, d; edge_sd(ei, e, s, d);
    atomicAdd(&s2[d], expf(leaky(a2s[s] + a2d[d]) - m2[d]));
}

// one thread per (edge, class)
__global__ void edge_acc2(const int* __restrict__ ei,
                          const float* __restrict__ a2s, const float* __restrict__ a2d,
                          const float* __restrict__ m2, const float* __restrict__ s2,
                          const float* __restrict__ g2, float* __restrict__ acc2) {
    long long i = (long long)blockIdx.x * blockDim.x + threadIdx.x;
    if (i >= (long long)E_TOT * CLS) return;
    int e = (int)(i >> 3);
    int c = (int)(i & 7);
    int s, d; edge_sd(ei, e, s, d);
    float alpha = expf(leaky(a2s[s] + a2d[d]) - m2[d]) / s2[d];
    atomicAdd(&acc2[d * CLS + c], alpha * g2[s * CLS + c]);
}

// ---------------- bias + log_softmax -> d_out ----------------
__global__ void final_logsoftmax(const float* __restrict__ acc2,
                                 const float* __restrict__ b2,
                                 float* __restrict__ out) {
    int n = blockIdx.x * blockDim.x + threadIdx.x;
    if (n >= N_NODES) return;
    float z[CLS], m = -INFINITY;
#pragma unroll
    for (int c = 0; c < CLS; ++c) { z[c] = acc2[n * CLS + c] + b2[c]; m = fmaxf(m, z[c]); }
    float sum = 0.f;
#pragma unroll
    for (int c = 0; c < CLS; ++c) sum += expf(z[c] - m);
    float lse = m + logf(sum);
#pragma unroll
    for (int c = 0; c < CLS; ++c) out[n * CLS + c] = z[c] - lse;
}

// ---------------- host launcher ----------------
extern "C" void kernel_launch(void* const* d_in, const int* in_sizes, int n_in,
                              void* d_out, int out_size, void* d_ws, size_t ws_size,
                              hipStream_t stream) {
    const float* x        = (const float*)d_in[0];
    const int*   ei       = (const int*)  d_in[1];   // [2, E]
    const float* W1       = (const float*)d_in[2];   // [256, 256]
    const float* att1_src = (const float*)d_in[3];   // [4, 64]
    const float* att1_dst = (const float*)d_in[4];
    const float* b1       = (const float*)d_in[5];   // [256]
    const float* W2       = (const float*)d_in[6];   // [256, 8]
    const float* att2_src = (const float*)d_in[7];   // [1, 8]
    const float* att2_dst = (const float*)d_in[8];
    const float* b2       = (const float*)d_in[9];   // [8]
    float* out = (float*)d_out;

    // workspace layout (floats); total = N*548 ~= 110 MB
    float* ws   = (float*)d_ws;
    float* h1   = ws;                              // N*C1
    float* acc1 = h1   + (size_t)N_NODES * C1;     // N*C1 (later: relu features)
    float* a1s  = acc1 + (size_t)N_NODES * C1;     // N*4
    float* a1d  = a1s  + (size_t)N_NODES * HEADS;
    float* m1   = a1d  + (size_t)N_NODES * HEADS;
    float* s1   = m1   + (size_t)N_NODES * HEADS;
    float* g2   = s1   + (size_t)N_NODES * HEADS;  // N*8
    float* a2s  = g2   + (size_t)N_NODES * CLS;    // N
    float* a2d  = a2s  + (size_t)N_NODES;
    float* m2   = a2d  + (size_t)N_NODES;
    float* s2   = m2   + (size_t)N_NODES;
    float* acc2 = s2   + (size_t)N_NODES;          // N*8

    const int B = 256;
    init_kernel<<<(N_NODES * C1 + B - 1) / B, B, 0, stream>>>(acc1, m1, s1, m2, s2, acc2);

    gemm1_wmma<<<dim3(N_NODES / 16, C1 / 64), 32, 0, stream>>>(x, W1, h1);
    alpha1_kernel<<<(N_NODES * HEADS + B - 1) / B, B, 0, stream>>>(h1, att1_src, att1_dst, a1s, a1d);

    edge_max1<<<(E_TOT + B - 1) / B, B, 0, stream>>>(ei, a1s, a1d, m1);
    edge_sum1<<<(E_TOT + B - 1) / B, B, 0, stream>>>(ei, a1s, a1d, m1, s1);
    edge_acc1<<<E_TOT, C1, 0, stream>>>(ei, a1s, a1d, m1, s1, h1, acc1);

    bias_relu<<<(N_NODES * C1 + B - 1) / B, B, 0, stream>>>(acc1, b1);

    gemm2_wmma<<<N_NODES / 16, 32, 0, stream>>>(acc1, W2, g2);
    alpha2_kernel<<<(N_NODES + B - 1) / B, B, 0, stream>>>(g2, att2_src, att2_dst, a2s, a2d);

    edge_max2<<<(E_TOT + B - 1) / B, B, 0, stream>>>(ei, a2s, a2d, m2);
    edge_sum2<<<(E_TOT + B - 1) / B, B, 0, stream>>>(ei, a2s, a2d, m2, s2);
    edge_acc2<<<((long long)E_TOT * CLS + B - 1) / B, B, 0, stream>>>(ei, a2s, a2d, m2, s2, g2, acc2);

    final_logsoftmax<<<(N_NODES + B - 1) / B, B, 0, stream>>>(acc2, b2, out);
}